// Graph_13778255085920
// MI455X (gfx1250) — compile-verified
//
#include <hip/hip_runtime.h>
#include <hip/hip_bf16.h>

#define EDGES_PER_WAVE 8
#define WAVES_PER_BLOCK 8

// ---------------------------------------------------------------------------
// COO SpMM scatter: out[dst] += val * x[src], D_FEAT = 64 floats = 256 B/row.
//
// wave32 layout: lane L owns columns {2L, 2L+1}  ->  one wave = one full row.
//   * 8 edges per wave; metadata base index forced wave-uniform with
//     readfirstlane -> metadata loads take the SMEM (scalar cache) path,
//     off the vector-memory path entirely.
//   * gather: global_load_b64 per lane (256 B contiguous per wave; x is
//     25.6 MB << 192 MB L2, so gathers are L2-resident after first touch).
//   * scatter: non-returning global_atomic_add_f32 resolved in L2 atomic units.
//   * global_prefetch_b8 on the streamed edge arrays ~16 KB ahead.
// ---------------------------------------------------------------------------
__global__ __launch_bounds__(256) void spmm_scatter(
    const float* __restrict__ x,
    const float* __restrict__ edge_val,
    const int*   __restrict__ edge_src,
    const int*   __restrict__ edge_dst,
    float*       __restrict__ out,
    int n_edges)
{
  const int lane   = threadIdx.x & 31;
  const int waveId = blockIdx.x * WAVES_PER_BLOCK + (threadIdx.x >> 5);

  int base = waveId * EDGES_PER_WAVE;
  base = __builtin_amdgcn_readfirstlane(base);   // force SGPR -> SMEM metadata
  if (base >= n_edges) return;

  const int colOff = lane << 1;                  // 2 consecutive floats per lane

  // Prefetch the edge streams ahead of consumption (global_prefetch_b8).
  if (lane == 0) {
    const int pf = base + 4096;
    if (pf < n_edges) {
      __builtin_prefetch(edge_src + pf, 0, 1);
      __builtin_prefetch(edge_dst + pf, 0, 1);
      __builtin_prefetch(edge_val + pf, 0, 1);
    }
  }

  if (base + EDGES_PER_WAVE <= n_edges) {
    // Wave-uniform metadata for 8 edges: wide scalar loads.
    const int4   sA = *(const int4*)  (edge_src + base);
    const int4   sB = *(const int4*)  (edge_src + base + 4);
    const int4   dA = *(const int4*)  (edge_dst + base);
    const int4   dB = *(const int4*)  (edge_dst + base + 4);
    const float4 vA = *(const float4*)(edge_val + base);
    const float4 vB = *(const float4*)(edge_val + base + 4);

    const int   s[EDGES_PER_WAVE] = {sA.x, sA.y, sA.z, sA.w, sB.x, sB.y, sB.z, sB.w};
    const int   d[EDGES_PER_WAVE] = {dA.x, dA.y, dA.z, dA.w, dB.x, dB.y, dB.z, dB.w};
    const float v[EDGES_PER_WAVE] = {vA.x, vA.y, vA.z, vA.w, vB.x, vB.y, vB.z, vB.w};

    // Phase 1: 8 independent gathers in flight (global_load_b64 each).
    float2 a[EDGES_PER_WAVE];
#pragma unroll
    for (int j = 0; j < EDGES_PER_WAVE; ++j) {
      a[j] = *(const float2*)(x + (((long)s[j]) << 6) + colOff);
    }

    // Phase 2: scale + non-returning f32 atomic scatter.
#pragma unroll
    for (int j = 0; j < EDGES_PER_WAVE; ++j) {
      float* p = out + (((long)d[j]) << 6) + colOff;
      unsafeAtomicAdd(p + 0, v[j] * a[j].x);
      unsafeAtomicAdd(p + 1, v[j] * a[j].y);
    }
  } else {
    // Tail: fewer than EDGES_PER_WAVE edges left for this wave.
#pragma clang loop unroll(disable) vectorize(disable)
    for (int e = base; e < n_edges; ++e) {
      const int   se = edge_src[e];
      const int   de = edge_dst[e];
      const float ve = edge_val[e];
      const float2 ae = *(const float2*)(x + (((long)se) << 6) + colOff);
      float* p = out + (((long)de) << 6) + colOff;
      unsafeAtomicAdd(p + 0, ve * ae.x);
      unsafeAtomicAdd(p + 1, ve * ae.y);
    }
  }
}

// ---------------------------------------------------------------------------
// Zero-fill the output (harness poisons d_out with 0xAA before timing).
// Vectorized 128-bit stores; minimal scalar tail.
// ---------------------------------------------------------------------------
__global__ __launch_bounds__(256) void spmm_zero(float* __restrict__ out, int n) {
  const long i = ((long)blockIdx.x * blockDim.x + threadIdx.x) << 2;
  if (i + 4 <= n) {
    *(float4*)(out + i) = float4{0.f, 0.f, 0.f, 0.f};
  } else if (i < n) {
#pragma clang loop unroll(disable) vectorize(disable)
    for (long j = i; j < n; ++j) out[j] = 0.f;
  }
}

// ---------------------------------------------------------------------------
// Inputs (setup_inputs order): x [N*64 f32], edge_val [E f32],
//                              edge_src [E i32], edge_dst [E i32]
// Output: out [N*64 f32]
// ---------------------------------------------------------------------------
extern "C" void kernel_launch(void* const* d_in, const int* in_sizes, int n_in,
                              void* d_out, int out_size, void* d_ws, size_t ws_size,
                              hipStream_t stream) {
  const float* x        = (const float*)d_in[0];
  const float* edge_val = (const float*)d_in[1];
  const int*   edge_src = (const int*)  d_in[2];
  const int*   edge_dst = (const int*)  d_in[3];
  float*       out      = (float*)d_out;

  const int n_edges = in_sizes[1];

  // 1) zero the output
  {
    const int nq     = (out_size + 3) / 4;
    const int blocks = (nq + 255) / 256;
    spmm_zero<<<blocks, 256, 0, stream>>>(out, out_size);
  }

  // 2) scatter-add
  {
    const int waves  = (n_edges + EDGES_PER_WAVE - 1) / EDGES_PER_WAVE;
    const int blocks = (waves + WAVES_PER_BLOCK - 1) / WAVES_PER_BLOCK;
    spmm_scatter<<<blocks, 256, 0, stream>>>(x, edge_val, edge_src, edge_dst,
                                             out, n_edges);
  }
}